// HybridModel_35699768164944
// MI455X (gfx1250) — compile-verified
//
#include <hip/hip_runtime.h>
#include <math.h>

typedef __attribute__((ext_vector_type(2))) float v2f;
typedef __attribute__((ext_vector_type(8))) float v8f;

#define NQ 6
#define NGATES 15
#define DFEAT 4096
#define ROWS_PER_BLOCK 128
#define THREADS 256

// Fused: h = x@W1 + b1 (WMMA f32 16x16x4), normalize, 15 RBS/Givens rotations,
// z = 1-2a^2 (NaN-guarded), sigmoid(z@W2 + b2).
__global__ __launch_bounds__(THREADS) void hybrid_pyramid_fused(
    const float* __restrict__ x,  const float* __restrict__ W1,
    const float* __restrict__ b1, const float* __restrict__ qw,
    const float* __restrict__ W2, const float* __restrict__ b2,
    float* __restrict__ out)
{
    __shared__ float w1s[DFEAT * NQ];          // 96 KB: W1 staged in LDS
    __shared__ float hS[ROWS_PER_BLOCK][NQ];   // 3 KB: per-row h values
    __shared__ float gsin[NGATES + 1];         // gate sines  (row-independent)
    __shared__ float gcos[NGATES + 1];         // gate cosines

    // Cooperative, coalesced stage of W1 (24576 floats = 6144 float4)
    {
        const float4* __restrict__ src = (const float4*)W1;
        float4* dst = (float4*)w1s;
        for (int i = threadIdx.x; i < (DFEAT * NQ) / 4; i += THREADS) dst[i] = src[i];
    }
    // Gate angles are shared by every row: 15 sincos per BLOCK, not per row.
    if (threadIdx.x < NGATES) {
        float w = qw[threadIdx.x];
        gsin[threadIdx.x] = sinf(w);
        gcos[threadIdx.x] = cosf(w);
    }
    __syncthreads();

    const int lane = threadIdx.x & 31;
    const int wave = threadIdx.x >> 5;
    const int m    = lane & 15;   // A: row in 16-row tile; B/D: column
    const int hi   = lane >> 4;   // half-wave select
    const int koff = hi * 2;      // K sub-offset per ISA 16x4 f32 A layout
    const int ncol = (m < NQ) ? m : (NQ - 1);  // clamp: cols 6..15 are don't-care

    const long rowBase = (long)blockIdx.x * ROWS_PER_BLOCK + (long)wave * 16;
    const float* __restrict__ xp = x + (rowBase + m) * DFEAT + koff;
    const float* __restrict__ bp = w1s + koff * NQ + ncol;

    v8f acc = {0.f, 0.f, 0.f, 0.f, 0.f, 0.f, 0.f, 0.f};

    #pragma unroll 8
    for (int kc = 0; kc < DFEAT; kc += 4) {
        // A fragment: lane m -> x[row+m][kc+koff .. kc+koff+1]  (8B aligned)
        v2f a = *(const v2f*)(xp + kc);
        // B fragment from LDS: W1[kc+koff][ncol], W1[kc+koff+1][ncol]
        v2f b;
        b.x = bp[kc * NQ];
        b.y = bp[kc * NQ + NQ];
        acc = __builtin_amdgcn_wmma_f32_16x16x4_f32(
            /*neg_a=*/false, a, /*neg_b=*/false, b,
            /*c_mod=*/(short)0, acc, /*reuse_a=*/false, /*reuse_b=*/false);
    }

    // D layout: VGPR j holds M=j (lanes 0-15) / M=j+8 (lanes 16-31), N=lane%16
    if (m < NQ) {
        #pragma unroll
        for (int j = 0; j < 8; ++j)
            hS[wave * 16 + hi * 8 + j][m] = acc[j];
    }
    __syncthreads();

    // One thread per row: tiny quantum-circuit tail (all compile-time unrolled,
    // no transcendentals except expf; gate sin/cos come from LDS).
    if (threadIdx.x < ROWS_PER_BLOCK) {
        const int r = threadIdx.x;
        float a[NQ];
        #pragma unroll
        for (int n = 0; n < NQ; ++n) a[n] = hS[r][n] + b1[n];

        float ss = 0.f;
        #pragma unroll
        for (int n = 0; n < NQ; ++n) ss += a[n] * a[n];
        float inv = 1.0f / sqrtf(ss);
        #pragma unroll
        for (int n = 0; n < NQ; ++n) a[n] *= inv;

        // pyramid top wires for n=6 (15 RBS gates)
        const int Q[NGATES] = {4, 3, 2, 4, 1, 3, 0, 2, 4, 1, 3, 2, 4, 3, 4};
        #pragma unroll
        for (int k = 0; k < NGATES; ++k) {
            const int q = Q[k];
            float s = gsin[k], c = gcos[k];
            float aq  =  c * a[q] + s * a[q + 1];
            float aq1 = -s * a[q] + c * a[q + 1];
            a[q] = aq; a[q + 1] = aq1;
        }

        float accum = b2[0];
        #pragma unroll
        for (int n = 0; n < NQ; ++n) {
            float z = 1.f - 2.f * a[n] * a[n];
            z = (z != z) ? 0.f : z;   // NaN guard (matches reference)
            accum += z * W2[n];
        }
        out[(long)blockIdx.x * ROWS_PER_BLOCK + r] = 1.f / (1.f + expf(-accum));
    }
}

extern "C" void kernel_launch(void* const* d_in, const int* in_sizes, int n_in,
                              void* d_out, int out_size, void* d_ws, size_t ws_size,
                              hipStream_t stream) {
    const float* x  = (const float*)d_in[0];
    const float* W1 = (const float*)d_in[1];
    const float* b1 = (const float*)d_in[2];
    const float* qw = (const float*)d_in[3];
    const float* W2 = (const float*)d_in[4];
    const float* b2 = (const float*)d_in[5];
    float* out = (float*)d_out;

    const int B = in_sizes[0] / DFEAT;        // 16384 rows
    dim3 grid(B / ROWS_PER_BLOCK);            // 128 blocks
    hybrid_pyramid_fused<<<grid, THREADS, 0, stream>>>(x, W1, b1, qw, W2, b2, out);
}